// QuantumGNN_59098749993604
// MI455X (gfx1250) — compile-verified
//
#include <hip/hip_runtime.h>
#include <math.h>

typedef __attribute__((ext_vector_type(2))) float v2f;
typedef __attribute__((ext_vector_type(8))) float v8f;

#define HID 64
#define PI_F 3.14159265358979323846f

// ---------------------------------------------------------------- degree prep
__global__ void k_init_deg(float* __restrict__ deg, int n) {
    int i = blockIdx.x * blockDim.x + threadIdx.x;
    if (i < n) deg[i] = 1.0f;              // self-loop contributes 1
}

__global__ void k_accum_deg(const int* __restrict__ dst, float* __restrict__ deg, int E) {
    int e = blockIdx.x * blockDim.x + threadIdx.x;
    if (e < E) unsafeAtomicAdd(&deg[dst[e]], 1.0f);   // global_atomic_add_f32, L2-resident
}

__global__ void k_rsqrt(const float* __restrict__ deg, float* __restrict__ dis, int n) {
    int i = blockIdx.x * blockDim.x + threadIdx.x;
    if (i < n) dis[i] = rsqrtf(deg[i]);
}

// ---------------------------------------------------------------- embedding
// x[i][j] = relu(noise[i][0..3] . W_emb[:,j] + b_emb[j])
__global__ void k_embed(const float* __restrict__ noise, const float* __restrict__ W,
                        const float* __restrict__ b, float* __restrict__ x, unsigned total) {
    unsigned gid = blockIdx.x * 256u + threadIdx.x;
    if (gid >= total) return;
    unsigned i = gid >> 6, j = gid & 63u;
    float4 p = ((const float4*)noise)[i];
    float s = p.x * W[0 * HID + j] + p.y * W[1 * HID + j] +
              p.z * W[2 * HID + j] + p.w * W[3 * HID + j] + b[j];
    x[gid] = fmaxf(s, 0.0f);
}

// ---------------------------------------------------------------- h = x @ W  (fp32 WMMA)
// One wave per 16-row slab of x; 4 N-tiles of 16 cols; K in steps of 4.
// A (16x4 f32): lanes 0-15 carry M=0..15 with {K0,K1}; lanes 16-31 carry {K2,K3}.
// B (4x16 f32): same mirrored layout; fed from LDS staged in a K-PAIRED layout:
//   Wp[(k>>1)*128 + 2*n + (k&1)] = W[k][n]
// so each B operand {W[kb][n], W[kb+1][n]} is one contiguous 8-byte LDS load.
// C/D (16x16 f32): 8 VGPRs; lanes 0-15 -> M=v, lanes 16-31 -> M=v+8, N=lane&15.
__global__ __launch_bounds__(256) void k_gemm64(const float* __restrict__ x,
                                                const float* __restrict__ W,
                                                float* __restrict__ h,
                                                int nTiles, int nRows) {
    __shared__ float Wp[HID * HID];
    for (int i = threadIdx.x; i < HID * HID; i += 256) {
        int k = i >> 6, n = i & 63;
        Wp[(k >> 1) * 128 + 2 * n + (k & 1)] = W[i];
    }
    __syncthreads();

    int wave = threadIdx.x >> 5;
    int lane = threadIdx.x & 31;
    int tile = blockIdx.x * 8 + wave;
    if (tile >= nTiles) return;                 // wave-uniform exit (EXEC all-1s for WMMA)

    int mlane = lane & 15;
    int koff  = (lane >> 4) * 2;                // 0 or 2
    int row   = tile * 16 + mlane;
    if (row >= nRows) row = nRows - 1;          // clamp (nRows is a multiple of 16 here)
    const float* xr = x + (size_t)row * HID;

    v8f a0 = {}, a1 = {}, a2 = {}, a3 = {};
    #pragma unroll
    for (int s = 0; s < 16; ++s) {
        int kb = 4 * s + koff;                  // even K base for this lane half
        v2f av = *(const v2f*)(xr + kb);
        const v2f* wrow = (const v2f*)&Wp[(kb >> 1) * 128];   // 64 paired v2f entries
        v2f b0 = wrow[ 0 + mlane];
        v2f b1 = wrow[16 + mlane];
        v2f b2 = wrow[32 + mlane];
        v2f b3 = wrow[48 + mlane];
        a0 = __builtin_amdgcn_wmma_f32_16x16x4_f32(false, av, false, b0, (short)0, a0, false, false);
        a1 = __builtin_amdgcn_wmma_f32_16x16x4_f32(false, av, false, b1, (short)0, a1, false, false);
        a2 = __builtin_amdgcn_wmma_f32_16x16x4_f32(false, av, false, b2, (short)0, a2, false, false);
        a3 = __builtin_amdgcn_wmma_f32_16x16x4_f32(false, av, false, b3, (short)0, a3, false, false);
    }

    int mbase = (lane >> 4) * 8;
    float* hb = h + (size_t)tile * 16 * HID;
    #pragma unroll
    for (int v = 0; v < 8; ++v) {
        float* r = hb + (size_t)(mbase + v) * HID + mlane;
        r[ 0] = a0[v];
        r[16] = a1[v];
        r[32] = a2[v];
        r[48] = a3[v];
    }
}

// ---------------------------------------------------------------- edge scatter (L2 atomics)
// agg[dst] += h[src] * dis[src]*dis[dst], one thread per (edge, feature)
__global__ void k_scatter(const int* __restrict__ src, const int* __restrict__ dst,
                          const float* __restrict__ dis, const float* __restrict__ h,
                          float* __restrict__ agg, unsigned total) {
    unsigned gid = blockIdx.x * 256u + threadIdx.x;
    if (gid >= total) return;
    unsigned e = gid >> 6, j = gid & 63u;
    int s = src[e], d = dst[e];
    float w = dis[s] * dis[d];
    unsafeAtomicAdd(&agg[(size_t)d * HID + j], h[(size_t)s * HID + j] * w);
}

// ---------------------------------------------------------------- finalize: self-loop + bias + relu
__global__ void k_finalize(const float* __restrict__ agg, const float* __restrict__ h,
                           const float* __restrict__ dis, const float* __restrict__ b,
                           float* __restrict__ x, unsigned total) {
    unsigned gid = blockIdx.x * 256u + threadIdx.x;
    if (gid >= total) return;
    unsigned i = gid >> 6, j = gid & 63u;
    float di = dis[i];
    float v = agg[gid] + h[gid] * (di * di) + b[j];
    x[gid] = fmaxf(v, 0.0f);
}

// ---------------------------------------------------------------- output head
__global__ void k_out(const float* __restrict__ x, const float* __restrict__ Wo,
                      const float* __restrict__ bo, float* __restrict__ out, int n) {
    int i = blockIdx.x * blockDim.x + threadIdx.x;
    if (i >= n) return;
    const float* xr = x + (size_t)i * HID;
    float a0 = bo[0], a1 = bo[1], a2 = bo[2];
    #pragma unroll 8
    for (int k = 0; k < HID; ++k) {
        float v = xr[k];
        a0 += v * Wo[k * 3 + 0];
        a1 += v * Wo[k * 3 + 1];
        a2 += v * Wo[k * 3 + 2];
    }
    out[i * 3 + 0] = tanhf(a0) * PI_F;
    out[i * 3 + 1] = tanhf(a1) * PI_F;
    out[i * 3 + 2] = tanhf(a2) * PI_F;
}

// ----------------------------------------------------------------
extern "C" void kernel_launch(void* const* d_in, const int* in_sizes, int n_in,
                              void* d_out, int out_size, void* d_ws, size_t ws_size,
                              hipStream_t stream) {
    const float* noise = (const float*)d_in[0];
    const int*   eidx  = (const int*)d_in[1];
    const float* W_emb = (const float*)d_in[2];
    const float* b_emb = (const float*)d_in[3];
    const float* Wl[3] = { (const float*)d_in[4], (const float*)d_in[6], (const float*)d_in[8] };
    const float* bl[3] = { (const float*)d_in[5], (const float*)d_in[7], (const float*)d_in[9] };
    const float* Wo = (const float*)d_in[10];
    const float* bo = (const float*)d_in[11];
    float* out = (float*)d_out;

    const int N = in_sizes[0] / 4;   // noise_params (1, N, 4)
    const int E = in_sizes[1] / 2;   // edge_index   (2, E)
    const int* src = eidx;
    const int* dst = eidx + E;

    // workspace carve-up (256B aligned): deg, dis, x, h, agg
    char* ws = (char*)d_ws;
    auto carve = [&](size_t bytes) { char* p = ws; ws += (bytes + 255) & ~(size_t)255; return p; };
    float* deg = (float*)carve((size_t)N * sizeof(float));
    float* dis = (float*)carve((size_t)N * sizeof(float));
    float* x   = (float*)carve((size_t)N * HID * sizeof(float));
    float* h   = (float*)carve((size_t)N * HID * sizeof(float));
    float* agg = (float*)carve((size_t)N * HID * sizeof(float));

    const unsigned nodeElems = (unsigned)N * HID;
    const unsigned edgeElems = (unsigned)E * HID;
    const int bN   = (N + 255) / 256;
    const int bE   = (E + 255) / 256;
    const int bNE  = (int)((nodeElems + 255u) / 256u);
    const int bEE  = (int)((edgeElems + 255u) / 256u);
    const int nTiles = (N + 15) / 16;
    const int bGemm  = (nTiles + 7) / 8;

    // degree + normalization
    k_init_deg<<<bN, 256, 0, stream>>>(deg, N);
    k_accum_deg<<<bE, 256, 0, stream>>>(dst, deg, E);
    k_rsqrt<<<bN, 256, 0, stream>>>(deg, dis, N);

    // embedding
    k_embed<<<bNE, 256, 0, stream>>>(noise, W_emb, b_emb, x, nodeElems);

    // three GCN layers
    for (int l = 0; l < 3; ++l) {
        hipMemsetAsync(agg, 0, (size_t)N * HID * sizeof(float), stream);
        k_gemm64<<<bGemm, 256, 0, stream>>>(x, Wl[l], h, nTiles, N);
        k_scatter<<<bEE, 256, 0, stream>>>(src, dst, dis, h, agg, edgeElems);
        k_finalize<<<bNE, 256, 0, stream>>>(agg, h, dis, bl[l], x, nodeElems);
    }

    // output head
    k_out<<<bN, 256, 0, stream>>>(x, Wo, bo, out, N);
}